// _GIN_56599079026905
// MI455X (gfx1250) — compile-verified
//
#include <hip/hip_runtime.h>

#define NFEAT   128
#define NHID    64
#define NNODES  50000
#define NEDGES  800000
#define NGRAPHS 512
#define NLAYERS 3
#define BN_EPS  1e-5f
#define NELEM   (NNODES * NHID)

typedef __attribute__((ext_vector_type(2))) float v2f;
typedef __attribute__((ext_vector_type(8))) float v8f;

__device__ __forceinline__ void atomAddF(float* p, float v) {
    __hip_atomic_fetch_add(p, v, __ATOMIC_RELAXED, __HIP_MEMORY_SCOPE_AGENT);
}

// ---------------- elementwise helpers ----------------
__global__ void k_fill(float* __restrict__ p, int n, float v) {
    int i = blockIdx.x * blockDim.x + threadIdx.x;
    if (i < n) p[i] = v;
}

__global__ void k_copy(const float* __restrict__ s, float* __restrict__ d, int n) {
    int i = blockIdx.x * blockDim.x + threadIdx.x;
    if (i < n) d[i] = s[i];
}

__global__ void k_count(const int* __restrict__ batch, float* __restrict__ counts, int n) {
    int i = blockIdx.x * blockDim.x + threadIdx.x;
    if (i < n) atomAddF(&counts[batch[i]], 1.0f);
}

// ---------------- WMMA GEMM: out[M x 64] = A[M x K] @ W[K x 64] + bias ----------------
// 128 threads = 4 waves; each wave computes a 16x64 tile via 4 f32 WMMA accumulators.
// W is staged in LDS as K-pairs: ldsW[kp*64 + c] = {W[2kp][c], W[2kp+1][c]}, so each
// B fragment is a single aligned ds_load_b64 into a consecutive VGPR pair.
template <int RELU, int K>
__global__ void k_gemm64(const float* __restrict__ A, const float* __restrict__ W,
                         const float* __restrict__ bias, float* __restrict__ out, int M) {
    __shared__ v2f ldsW[(NFEAT / 2) * NHID];   // up to 32 KB; (K/2)*64 used
    const int tid = threadIdx.x;
    for (int i = tid; i < (K / 2) * NHID; i += 128) {
        const int kp = i >> 6, c = i & 63;
        v2f w;
        w.x = W[(2 * kp) * NHID + c];
        w.y = W[(2 * kp + 1) * NHID + c];
        ldsW[i] = w;
    }
    __syncthreads();

    const int wave = tid >> 5;
    const int lane = tid & 31;
    const int hi   = lane >> 4;      // lanes 0-15: K=k0,k0+1   lanes 16-31: K=k0+2,k0+3
    const int lm   = lane & 15;
    const int rowBase = blockIdx.x * 64 + wave * 16;

    const int   arow = rowBase + lm;
    const bool  ok   = arow < M;
    const float mask = ok ? 1.0f : 0.0f;
    const v2f*  Ap   = (const v2f*)(A + (size_t)(ok ? arow : 0) * K);

    v8f acc0 = {}, acc1 = {}, acc2 = {}, acc3 = {};
#pragma unroll
    for (int k0 = 0; k0 < K; k0 += 4) {
        const int kp = (k0 >> 1) + hi;         // pair index for this lane half
        v2f a = Ap[kp];                        // unconditional global_load_b64
        a.x *= mask;
        a.y *= mask;
        const v2f* wp = &ldsW[kp * NHID];
        v2f b0 = wp[lm];
        v2f b1 = wp[16 + lm];
        v2f b2 = wp[32 + lm];
        v2f b3 = wp[48 + lm];
        acc0 = __builtin_amdgcn_wmma_f32_16x16x4_f32(false, a, false, b0, (short)0, acc0, false, false);
        acc1 = __builtin_amdgcn_wmma_f32_16x16x4_f32(false, a, false, b1, (short)0, acc1, false, false);
        acc2 = __builtin_amdgcn_wmma_f32_16x16x4_f32(false, a, false, b2, (short)0, acc2, false, false);
        acc3 = __builtin_amdgcn_wmma_f32_16x16x4_f32(false, a, false, b3, (short)0, acc3, false, false);
    }

    const float bv0 = bias[lm];
    const float bv1 = bias[16 + lm];
    const float bv2 = bias[32 + lm];
    const float bv3 = bias[48 + lm];

#pragma unroll
    for (int r = 0; r < 8; ++r) {
        const int orow = rowBase + r + 8 * hi;   // C/D layout: lanes 0-15 -> M=r, 16-31 -> M=r+8
        if (orow < M) {
            float* op = out + (size_t)orow * NHID;
            float v0 = acc0[r] + bv0;
            float v1 = acc1[r] + bv1;
            float v2 = acc2[r] + bv2;
            float v3 = acc3[r] + bv3;
            if (RELU) {
                v0 = fmaxf(v0, 0.0f); v1 = fmaxf(v1, 0.0f);
                v2 = fmaxf(v2, 0.0f); v3 = fmaxf(v3, 0.0f);
            }
            op[lm]      = v0;
            op[16 + lm] = v1;
            op[32 + lm] = v2;
            op[48 + lm] = v3;
        }
    }
}

// ---------------- BatchNorm (training-mode batch stats) ----------------
// stats layout: [0,64)=sum  [64,128)=sumsq  [128,192)=scale  [192,256)=shift
__global__ void k_bn_stats(const float* __restrict__ y, int M, float* __restrict__ stats) {
    __shared__ float ss[256];
    __shared__ float sq[256];
    const int tid = threadIdx.x;
    const int c = tid & 63, q = tid >> 6;
    float s = 0.0f, s2 = 0.0f;
    for (int r = blockIdx.x * 4 + q; r < M; r += gridDim.x * 4) {
        const float v = y[(size_t)r * NHID + c];
        s += v; s2 += v * v;
    }
    ss[tid] = s; sq[tid] = s2;
    __syncthreads();
    if (q == 0) {
        s  = ss[c] + ss[c + 64] + ss[c + 128] + ss[c + 192];
        s2 = sq[c] + sq[c + 64] + sq[c + 128] + sq[c + 192];
        atomAddF(&stats[c], s);
        atomAddF(&stats[64 + c], s2);
    }
}

__global__ void k_bn_finalize(float* __restrict__ stats, int M,
                              const float* __restrict__ gamma, const float* __restrict__ beta) {
    const int c = threadIdx.x;
    if (c < 64) {
        const float invM = 1.0f / (float)M;
        const float mean = stats[c] * invM;
        const float var  = stats[64 + c] * invM - mean * mean;
        const float sc   = gamma[c] * rsqrtf(var + BN_EPS);
        stats[128 + c] = sc;
        stats[192 + c] = beta[c] - mean * sc;
    }
}

__global__ void k_bn_apply(const float* __restrict__ y, const float* __restrict__ stats,
                           float* __restrict__ h, int n, int relu) {
    int i = blockIdx.x * blockDim.x + threadIdx.x;
    if (i < n) {
        const int c = i & 63;
        float v = y[i] * stats[128 + c] + stats[192 + c];
        if (relu) v = fmaxf(v, 0.0f);
        h[i] = v;
    }
}

// ---------------- edge aggregation: z[dst] += h[src] (one thread per edge,col) ----------------
__global__ void k_edge(const float* __restrict__ h, const int* __restrict__ src,
                       const int* __restrict__ dst, float* __restrict__ z) {
    int i = blockIdx.x * blockDim.x + threadIdx.x;
    if (i < NEDGES * NHID) {
        const int e = i >> 6, c = i & 63;
        atomAddF(&z[(size_t)dst[e] * NHID + c], h[(size_t)src[e] * NHID + c]);
    }
}

// ---------------- global mean pool ----------------
__global__ void k_pool_accum(const float* __restrict__ h, const int* __restrict__ batch,
                             float* __restrict__ dst) {
    int i = blockIdx.x * blockDim.x + threadIdx.x;
    if (i < NELEM) {
        const int row = i >> 6, c = i & 63;
        atomAddF(&dst[(size_t)batch[row] * NHID + c], h[i]);
    }
}

__global__ void k_pool_div(float* __restrict__ dst, const float* __restrict__ counts) {
    int i = blockIdx.x * blockDim.x + threadIdx.x;
    if (i < NGRAPHS * NHID) {
        dst[i] = dst[i] / fmaxf(counts[i >> 6], 1.0f);
    }
}

// ---------------- host orchestration ----------------
extern "C" void kernel_launch(void* const* d_in, const int* in_sizes, int n_in,
                              void* d_out, int out_size, void* d_ws, size_t ws_size,
                              hipStream_t stream) {
    const float* x    = (const float*)d_in[0];
    const int*   eidx = (const int*)d_in[1];
    const int*   bat  = (const int*)d_in[2];
    const float* Wt   = (const float*)d_in[3];
    const float* bt   = (const float*)d_in[4];
    const float* gt   = (const float*)d_in[5];
    const float* bet  = (const float*)d_in[6];
    const float* W1   = (const float*)d_in[7];
    const float* b1   = (const float*)d_in[8];
    const float* W2   = (const float*)d_in[9];
    const float* b2   = (const float*)d_in[10];
    const float* g    = (const float*)d_in[11];
    const float* be   = (const float*)d_in[12];
    float* out = (float*)d_out;

    float* ws     = (float*)d_ws;
    float* bufH   = ws;                     // node features (post-BN)
    float* bufZ   = ws + (size_t)NELEM;     // pre-BN / z buffer
    float* bufT   = ws + (size_t)2 * NELEM; // MLP hidden
    float* stats  = ws + (size_t)3 * NELEM; // 256 floats
    float* counts = stats + 256;            // 512 floats

    const int T = 256;
    const int gElem  = (NELEM + T - 1) / T;
    const int gEdge  = (NEDGES * NHID + T - 1) / T;
    const int gPool  = (NGRAPHS * NHID + T - 1) / T;
    const int gNodes = (NNODES + T - 1) / T;
    const int gGemm  = (NNODES + 63) / 64;

    const int* src = eidx;            // edge_index[0]
    const int* dst = eidx + NEDGES;   // edge_index[1]

    // per-graph node counts (reused for all 4 embeds)
    k_fill<<<(NGRAPHS + T - 1) / T, T, 0, stream>>>(counts, NGRAPHS, 0.0f);
    k_count<<<gNodes, T, 0, stream>>>(bat, counts, NNODES);

    // input transform: bufZ = x @ Wt + bt ; bufH = BN(bufZ)
    k_gemm64<0, NFEAT><<<gGemm, 128, 0, stream>>>(x, Wt, bt, bufZ, NNODES);
    k_fill<<<1, 128, 0, stream>>>(stats, 128, 0.0f);
    k_bn_stats<<<256, 256, 0, stream>>>(bufZ, NNODES, stats);
    k_bn_finalize<<<1, 64, 0, stream>>>(stats, NNODES, gt, bet);
    k_bn_apply<<<gElem, T, 0, stream>>>(bufZ, stats, bufH, NELEM, 0);

    // embed 0
    {
        float* o0 = out;
        k_fill<<<gPool, T, 0, stream>>>(o0, NGRAPHS * NHID, 0.0f);
        k_pool_accum<<<gElem, T, 0, stream>>>(bufH, bat, o0);
        k_pool_div<<<gPool, T, 0, stream>>>(o0, counts);
    }

    for (int l = 0; l < NLAYERS; ++l) {
        // z = h + sum_{j->i} h_j
        k_copy<<<gElem, T, 0, stream>>>(bufH, bufZ, NELEM);
        k_edge<<<gEdge, T, 0, stream>>>(bufH, src, dst, bufZ);

        // MLP: t = relu(z @ W1 + b1) ; z = t @ W2 + b2
        k_gemm64<1, NHID><<<gGemm, 128, 0, stream>>>(bufZ, W1 + (size_t)l * NHID * NHID,
                                                     b1 + (size_t)l * NHID, bufT, NNODES);
        k_gemm64<0, NHID><<<gGemm, 128, 0, stream>>>(bufT, W2 + (size_t)l * NHID * NHID,
                                                     b2 + (size_t)l * NHID, bufZ, NNODES);

        // h = relu(BN(z))
        k_fill<<<1, 128, 0, stream>>>(stats, 128, 0.0f);
        k_bn_stats<<<256, 256, 0, stream>>>(bufZ, NNODES, stats);
        k_bn_finalize<<<1, 64, 0, stream>>>(stats, NNODES, g + (size_t)l * NHID, be + (size_t)l * NHID);
        k_bn_apply<<<gElem, T, 0, stream>>>(bufZ, stats, bufH, NELEM, 1);

        // embed l+1
        float* oe = out + (size_t)(l + 1) * NGRAPHS * NHID;
        k_fill<<<gPool, T, 0, stream>>>(oe, NGRAPHS * NHID, 0.0f);
        k_pool_accum<<<gElem, T, 0, stream>>>(bufH, bat, oe);
        k_pool_div<<<gPool, T, 0, stream>>>(oe, counts);
    }
}